// RFIconv_13529146983228
// MI455X (gfx1250) — compile-verified
//
#include <hip/hip_runtime.h>
#include <stdint.h>

// ---------------------------------------------------------------------------
// Stage 1: a = heaviside(conv3x3(x,w3)) [2ch], b = heaviside(conv5x5(x,w5))
//          [4ch] via V_WMMA_F32_16X16X4_F32 (exact fp32), bit-packed to ws.
// Stage 2: c = dilate5x5(a); d = b & ~repeat(c,2); out = (sum conv(a,w3b) +
//          sum conv(d,w5b)) > 1e-7 -> word-parallel bit logic + LDS LUT sums.
// ---------------------------------------------------------------------------

typedef float v2f __attribute__((ext_vector_type(2)));
typedef float v8f __attribute__((ext_vector_type(8)));

#define B_   8
#define H_   2048
#define W_   1024
#define WW_  (W_/32)            // 32 words of packed bits per row

#define A_WORDS (B_*2*H_*WW_)   // a bitmask words (4 MB)
#define B_WORDS (B_*4*H_*WW_)   // b bitmask words (8 MB)

__device__ __forceinline__ int bperm_i(int idx, int v) {
  return __builtin_amdgcn_ds_bpermute(idx, v);
}
__device__ __forceinline__ float bperm_f(int idx, float v) {
  return __int_as_float(__builtin_amdgcn_ds_bpermute(idx, __float_as_int(v)));
}

// ---------------------------------------------------------------------------
// Stage 1. One wave = 256 consecutive pixels of one row (16 M-tiles of 16 px).
// Per tile: M = 16 pixels, N = 16 (ch0..1 = w3 embedded, ch2..5 = w5),
// K = 40 slots (5 kernel rows x 8, slots 5..7 zero) -> 10 chained f32 WMMAs.
// A fragment fetched by ds_bpermute from a 20-lane row window; interior tiles
// use unpredicated loads (all 32 lanes in-bounds), edges take the slow path.
// ---------------------------------------------------------------------------
#define SEG 256

__global__ __launch_bounds__(256) void stage1_wmma(
    const float* __restrict__ x, const float* __restrict__ w3,
    const float* __restrict__ w5, uint32_t* __restrict__ aB,
    uint32_t* __restrict__ bB) {
  __shared__ float Bmat[16][40];   // zero-padded B matrix (N x Kslots)

  const int tid   = threadIdx.x;
  const int lane  = tid & 31;
  const int waveG = blockIdx.x * (blockDim.x >> 5) + (tid >> 5);
  const int segs  = W_ / SEG;                 // 4
  int tmp = waveG;
  const int seg = tmp % segs; tmp /= segs;
  const int y   = tmp % H_;   tmp /= H_;
  const int bi  = tmp;                        // batch
  const int xseg = seg * SEG;

  // ---- build padded B matrix once per block ------------------------------
  for (int i = tid; i < 16 * 40; i += 256) {
    const int n = i / 40, k = i % 40, r = k >> 3, s = k & 7;
    float v = 0.f;
    if (n < 2) {                       // 3x3 kernel embedded at rows/cols 1..3
      if (r >= 1 && r <= 3 && s >= 1 && s <= 3)
        v = w3[n * 9 + (r - 1) * 3 + (s - 1)];
    } else if (n < 6) {                // 5x5 kernel, cols 0..4
      if (s < 5) v = w5[(n - 2) * 25 + r * 5 + s];
    }
    Bmat[n][k] = v;
  }
  __syncthreads();

  const int n     = lane & 15;                // N index (channel)
  const int m     = lane & 15;                // M index (pixel-in-tile)
  const int sbase = (lane >> 4) << 1;         // 0 / 2

  float bw0[10], bw1[10];
#pragma unroll
  for (int c = 0; c < 10; ++c) {
    const int k = (c >> 1) * 8 + (c & 1) * 4 + sbase;
    bw0[c] = Bmat[n][k];
    bw1[c] = Bmat[n][k + 1];
  }

  // bpermute byte-indices (source lane = m + s)
  const int i0h0 = (m + sbase) * 4;
  const int i1h0 = i0h0 + 4;
  const int i0h1 = i0h0 + 16;
  const int i1h1 = i0h0 + 20;

  const bool rowsIn = (y >= 2) && (y + 2 < H_);
  const float* xrow = x + ((size_t)bi * H_ + y) * W_;   // row y base

  uint32_t prev16 = 0;
#pragma unroll 2
  for (int t = 0; t < SEG / 16; ++t) {
    const int x0  = xseg + t * 16;
    const int col = x0 - 2 + lane;
    float xr[5];
    if (rowsIn && x0 >= 16 && x0 + 18 <= W_) {
      // interior: every lane in-bounds (col in [14, 1021]); lanes 20..31 feed
      // only zero-weight B slots, so their (valid) values are don't-cares.
      const float* p = xrow + (col - 2 * W_);
#pragma unroll
      for (int r = 0; r < 5; ++r) xr[r] = p[r * W_];
    } else {
      const bool cok = (lane < 20) && (col >= 0) && (col < W_);
#pragma unroll
      for (int r = 0; r < 5; ++r) {
        const int yr = y + r - 2;
        float v = 0.f;
        if (cok && yr >= 0 && yr < H_) v = xrow[(r - 2) * W_ + col];
        xr[r] = v;
      }
    }

    // ---- 10 chained fp32 WMMAs ------------------------------------------
    v8f acc = {0.f, 0.f, 0.f, 0.f, 0.f, 0.f, 0.f, 0.f};
#pragma unroll
    for (int c = 0; c < 10; ++c) {
      const int r = c >> 1, h = c & 1;
      v2f A, Bv;
      A.x  = bperm_f(h ? i0h1 : i0h0, xr[r]);
      A.y  = bperm_f(h ? i1h1 : i1h0, xr[r]);
      Bv.x = bw0[c];
      Bv.y = bw1[c];
      acc = __builtin_amdgcn_wmma_f32_16x16x4_f32(
          false, A, false, Bv, (short)0, acc, false, false);
    }

    // ---- heaviside + pack: lane holds 8 pixels of channel n --------------
    uint32_t m8 = 0;
#pragma unroll
    for (int v = 0; v < 8; ++v)
      m8 |= (acc[v] > 0.f) ? (1u << v) : 0u;
    const uint32_t hi8 = (uint32_t)bperm_i(((lane + 16) & 31) * 4, (int)m8);
    const uint32_t m16 = (m8 & 0xffu) | ((hi8 & 0xffu) << 8);

    if ((t & 1) == 0) {
      prev16 = m16;
    } else {
      const uint32_t word = prev16 | (m16 << 16);
      const int w = x0 >> 5;
      if (lane < 2)
        aB[(((size_t)bi * 2 + lane) * H_ + y) * WW_ + w] = word;
      else if (lane < 6)
        bB[(((size_t)bi * 4 + (lane - 2)) * H_ + y) * WW_ + w] = word;
    }
  }
}

// ---------------------------------------------------------------------------
// Stage 2: tile = 16 rows x 256 cols per 256-thread block. Word-parallel bit
// logic in LDS, then per-pixel sums via 32-entry / 8-entry partial-sum LUTs
// (binary conv == masked weight sum -> one LDS lookup per (ch, kernel-row)).
// ---------------------------------------------------------------------------
#define TY 16
#define TX 256

__global__ __launch_bounds__(256) void stage2_bits(
    const uint32_t* __restrict__ aB, const uint32_t* __restrict__ bB,
    const float* __restrict__ w3b, const float* __restrict__ w5b,
    float* __restrict__ out) {
  __shared__ uint32_t aw[2][TY + 8][12];  // a bits, rows y0-4..y0+19, w0-2..w0+9
  __shared__ uint32_t hd[2][TY + 8][10];  // horiz dilation of a, w0-1..w0+8
  __shared__ uint32_t db[4][TY + 4][10];  // b bits -> overwritten with d bits
  __shared__ float lut5[4][5][32];        // 5-bit masked partial sums of w5b
  __shared__ float lut3[2][3][8];         // 3-bit masked partial sums of w3b

  const int tid = threadIdx.x;
  int tmp = blockIdx.x;
  const int sx = tmp % (W_ / TX); tmp /= (W_ / TX);
  const int sy = tmp % (H_ / TY); tmp /= (H_ / TY);
  const int bi = tmp;
  const int x0 = sx * TX, y0 = sy * TY;
  const int w0 = x0 >> 5;

  // ---- cooperative fills (OOB -> 0 implements zero padding) --------------
  for (int i = tid; i < 2 * 24 * 12; i += 256) {
    const int wi = i % 12, r = (i / 12) % 24, ch = i / (12 * 24);
    const int gw = w0 - 2 + wi, gy = y0 - 4 + r;
    uint32_t v = 0;
    if (gw >= 0 && gw < WW_ && gy >= 0 && gy < H_)
      v = aB[(((size_t)bi * 2 + ch) * H_ + gy) * WW_ + gw];
    aw[ch][r][wi] = v;
  }
  for (int i = tid; i < 4 * 20 * 10; i += 256) {
    const int wi = i % 10, r = (i / 10) % 20, ch = i / (10 * 20);
    const int gw = w0 - 1 + wi, gy = y0 - 2 + r;
    uint32_t v = 0;
    if (gw >= 0 && gw < WW_ && gy >= 0 && gy < H_)
      v = bB[(((size_t)bi * 4 + ch) * H_ + gy) * WW_ + gw];
    db[ch][r][wi] = v;
  }
  // ---- masked-sum LUTs ----------------------------------------------------
  for (int i = tid; i < 4 * 5 * 32; i += 256) {
    const int t = i & 31, j = (i >> 5) % 5, ch = i / 160;
    float s = 0.f;
#pragma unroll
    for (int dx = 0; dx < 5; ++dx)
      if (t & (1 << dx)) s += w5b[ch * 25 + j * 5 + dx];
    lut5[ch][j][t] = s;
  }
  if (tid < 48) {
    const int t = tid & 7, j = (tid >> 3) % 3, ch = tid / 24;
    float s = 0.f;
#pragma unroll
    for (int dx = 0; dx < 3; ++dx)
      if (t & (1 << dx)) s += w3b[ch * 9 + j * 3 + dx];
    lut3[ch][j][t] = s;
  }
  __syncthreads();

  // ---- horizontal +/-2 dilation (word funnel shifts) ----------------------
  for (int i = tid; i < 2 * 24 * 10; i += 256) {
    const int wi = i % 10, r = (i / 10) % 24, ch = i / (10 * 24);
    const uint32_t L = aw[ch][r][wi], M = aw[ch][r][wi + 1], R = aw[ch][r][wi + 2];
    const uint64_t lo = ((uint64_t)M << 32) | L;
    const uint64_t hi = ((uint64_t)R << 32) | M;
    hd[ch][r][wi] = M | (uint32_t)(lo >> 31) | (uint32_t)(lo >> 30) |
                        (uint32_t)(hi >> 1)  | (uint32_t)(hi >> 2);
  }
  __syncthreads();

  // ---- d = b & ~c, c = vertical OR of 5 dilated rows ----------------------
  for (int i = tid; i < 4 * 20 * 10; i += 256) {
    const int wi = i % 10, r = (i / 10) % 20, ch = i / (10 * 20);
    const int ac = ch >> 1;   // repeat(c, 2) channel mapping
    const uint32_t c = hd[ac][r][wi] | hd[ac][r + 1][wi] | hd[ac][r + 2][wi] |
                       hd[ac][r + 3][wi] | hd[ac][r + 4][wi];
    db[ch][r][wi] &= ~c;
  }
  __syncthreads();

  // ---- per-pixel masked weight sums (sliding row windows) ----------------
  const int xl = tid;
  const int xg = x0 + xl;
  const int bp = xl & 31;
  const int iwD = (xl >> 5) + 1;   // central word in 10-wide arrays
  const int iwA = (xl >> 5) + 2;   // central word in 12-wide arrays

  auto row3 = [&](int ch, int r) -> uint32_t {   // bit k = pixel x-1+k
    const uint32_t p = aw[ch][r][iwA - 1], cu = aw[ch][r][iwA], nx = aw[ch][r][iwA + 1];
    const uint64_t mm = ((((uint64_t)nx << 32) | cu) << 1) | (p >> 31);
    return (uint32_t)(mm >> bp) & 7u;
  };
  auto row5 = [&](int ch, int r) -> uint32_t {   // bit k = pixel x-2+k
    const uint32_t p = db[ch][r][iwD - 1], cu = db[ch][r][iwD], nx = db[ch][r][iwD + 1];
    const uint64_t mm = ((((uint64_t)nx << 32) | cu) << 2) | (p >> 30);
    return (uint32_t)(mm >> bp) & 31u;
  };

  uint32_t a3[2][3];
  uint32_t d5[4][5];
#pragma unroll
  for (int ch = 0; ch < 2; ++ch) {
    a3[ch][0] = row3(ch, 3);
    a3[ch][1] = row3(ch, 4);
  }
#pragma unroll
  for (int ch = 0; ch < 4; ++ch)
#pragma unroll
    for (int j = 0; j < 4; ++j) d5[ch][j] = row5(ch, j);

  for (int k = 0; k < TY; ++k) {
    const int yy = y0 + k;
#pragma unroll
    for (int ch = 0; ch < 2; ++ch) a3[ch][2] = row3(ch, k + 5);
#pragma unroll
    for (int ch = 0; ch < 4; ++ch) d5[ch][4] = row5(ch, k + 4);

    float acc = 0.f;
#pragma unroll
    for (int ch = 0; ch < 2; ++ch)
#pragma unroll
      for (int j = 0; j < 3; ++j) acc += lut3[ch][j][a3[ch][j]];
#pragma unroll
    for (int ch = 0; ch < 4; ++ch)
#pragma unroll
      for (int j = 0; j < 5; ++j) acc += lut5[ch][j][d5[ch][j]];

    out[((size_t)bi * H_ + yy) * W_ + xg] = (acc > 1e-7f) ? 1.f : 0.f;

#pragma unroll
    for (int ch = 0; ch < 2; ++ch) {
      a3[ch][0] = a3[ch][1];
      a3[ch][1] = a3[ch][2];
    }
#pragma unroll
    for (int ch = 0; ch < 4; ++ch)
#pragma unroll
      for (int j = 0; j < 4; ++j) d5[ch][j] = d5[ch][j + 1];
  }
}

// ---------------------------------------------------------------------------
extern "C" void kernel_launch(void* const* d_in, const int* in_sizes, int n_in,
                              void* d_out, int out_size, void* d_ws,
                              size_t ws_size, hipStream_t stream) {
  (void)in_sizes; (void)n_in; (void)out_size; (void)ws_size;
  const float* x   = (const float*)d_in[0];
  const float* w3  = (const float*)d_in[1];
  const float* w5  = (const float*)d_in[2];
  const float* w3b = (const float*)d_in[3];
  const float* w5b = (const float*)d_in[4];
  float* out = (float*)d_out;

  uint32_t* aB = (uint32_t*)d_ws;            // 4 MB
  uint32_t* bB = aB + A_WORDS;               // 8 MB (ws needs >= 12 MB)

  const dim3 blk(256);
  const int waves1 = B_ * H_ * (W_ / SEG);   // 65536 waves
  stage1_wmma<<<dim3(waves1 / 8), blk, 0, stream>>>(x, w3, w5, aB, bB);

  const int blocks2 = B_ * (H_ / TY) * (W_ / TX);  // 4096
  stage2_bits<<<dim3(blocks2), blk, 0, stream>>>(aB, bB, w3b, w5b, out);
}